// GNNActor_48653389529335
// MI455X (gfx1250) — compile-verified
//
#include <hip/hip_runtime.h>
#include <hip/hip_bf16.h>

typedef float v2f __attribute__((ext_vector_type(2)));
typedef float v8f __attribute__((ext_vector_type(8)));

// ---------------- float <-> orderable-uint for atomic segment-max ----------
__device__ __forceinline__ unsigned enc_f(float x) {
    unsigned b = __float_as_uint(x);
    return (b & 0x80000000u) ? ~b : (b | 0x80000000u);
}
__device__ __forceinline__ float dec_f(unsigned k) {
    unsigned b = (k & 0x80000000u) ? (k & 0x7FFFFFFFu) : ~k;
    return __uint_as_float(b);
}

// ---------------- self-loop edge_attr mean (fill_value='mean') -------------
__global__ void ea_stats_kernel(const int* __restrict__ dst,
                                const float* __restrict__ ea,
                                float* __restrict__ cnt,
                                float* __restrict__ easum, int E) {
    int e = blockIdx.x * blockDim.x + threadIdx.x;
    if (e >= E) return;
    int d = dst[e];
    atomicAdd(&cnt[d], 1.0f);
    atomicAdd(&easum[2 * d + 0], ea[2 * e + 0]);
    atomicAdd(&easum[2 * d + 1], ea[2 * e + 1]);
}

__global__ void ea_div_kernel(float* __restrict__ easum,
                              const float* __restrict__ cnt, int N) {
    int i = blockIdx.x * blockDim.x + threadIdx.x;
    if (i >= N) return;
    float inv = 1.0f / fmaxf(cnt[i], 1.0f);
    easum[2 * i + 0] *= inv;
    easum[2 * i + 1] *= inv;
}

// ---------------- fp32 WMMA GEMM: Hout[N,128] = X[N,K] @ W[K,128] ----------
// One wave computes one 16x16 output tile via V_WMMA_F32_16X16X4_F32.
// Block = 256 threads = 8 waves = the 8 column tiles of H=128.
template <int K>
__global__ __launch_bounds__(256) void gemm_wmma_kernel(
    const float* __restrict__ X, const float* __restrict__ W,
    float* __restrict__ Hout, int N) {
    const int wave = threadIdx.x >> 5;   // 0..7 -> column tile
    const int lane = threadIdx.x & 31;
    const int half = lane >> 4;          // 0: lanes 0-15, 1: lanes 16-31
    const int l15  = lane & 15;

    int row = blockIdx.x * 16 + l15;     // A row (M)
    if (row > N - 1) row = N - 1;        // clamp: keep EXEC all-ones for WMMA
    const int colB = wave * 16 + l15;    // B column (N)

    v8f c = {};
#pragma unroll
    for (int k = 0; k < K; k += 4) {
        // A 16x4 layout: VGPR0 = K(k+0) lanes0-15 / K(k+2) lanes16-31;
        //                VGPR1 = K(k+1) / K(k+3)
        const int ka = k + half * 2;
        v2f a;
        a.x = X[(size_t)row * K + ka + 0];
        a.y = X[(size_t)row * K + ka + 1];
        // B 4x16 layout mirrors A: lane half selects K pair, lane&15 = N
        v2f b;
        b.x = W[(size_t)(ka + 0) * 128 + colB];
        b.y = W[(size_t)(ka + 1) * 128 + colB];
        c = __builtin_amdgcn_wmma_f32_16x16x4_f32(
                /*neg_a=*/false, a, /*neg_b=*/false, b,
                /*c_mod=*/(short)0, c, /*reuse_a=*/false, /*reuse_b=*/false);
    }
    // C/D layout: VGPR r: lanes0-15 -> M=r, lanes16-31 -> M=8+r; N = lane&15
    const int m0 = blockIdx.x * 16 + half * 8;
    const int n0 = wave * 16 + l15;
#pragma unroll
    for (int r = 0; r < 8; ++r) {
        int m = m0 + r;
        if (m < N) Hout[(size_t)m * 128 + n0] = c[r];
    }
}

// ---------------- per-node attention scalars s=h.a_s, d=h.a_d --------------
__global__ void node_scalars_kernel(const float* __restrict__ h,
                                    const float* __restrict__ a_s,
                                    const float* __restrict__ a_d,
                                    float* __restrict__ s,
                                    float* __restrict__ dsc, int N) {
    int wid = (blockIdx.x * blockDim.x + threadIdx.x) >> 5;
    int lane = threadIdx.x & 31;
    if (wid >= N) return;
    float ss = 0.f, dd = 0.f;
    const float* hr = h + (size_t)wid * 128;
#pragma unroll
    for (int j = lane; j < 128; j += 32) {
        float v = hr[j];
        ss += v * a_s[j];
        dd += v * a_d[j];
    }
#pragma unroll
    for (int o = 16; o > 0; o >>= 1) {
        ss += __shfl_down(ss, o);
        dd += __shfl_down(dd, o);
    }
    if (lane == 0) { s[wid] = ss; dsc[wid] = dd; }
}

// ---------------- c2 = We @ a_e  (2-vector)  -------------------------------
__global__ void compute_c2_kernel(const float* __restrict__ We,
                                  const float* __restrict__ a_e,
                                  float* __restrict__ c2) {
    int w = threadIdx.x >> 5;        // 0..1
    int lane = threadIdx.x & 31;
    float acc = 0.f;
#pragma unroll
    for (int j = lane; j < 128; j += 32) acc += We[w * 128 + j] * a_e[j];
#pragma unroll
    for (int o = 16; o > 0; o >>= 1) acc += __shfl_down(acc, o);
    if (lane == 0) c2[w] = acc;
}

// ---------------- pass 1: logits + segment max -----------------------------
__global__ void edge_logits_max_kernel(const int* __restrict__ src,
                                       const int* __restrict__ dst,
                                       const float* __restrict__ ea,
                                       const float* __restrict__ ea_mean,
                                       const float* __restrict__ c2,
                                       const float* __restrict__ s,
                                       const float* __restrict__ dsc,
                                       float* __restrict__ logits,
                                       unsigned* __restrict__ mkey,
                                       int E, int EE) {
    int e = blockIdx.x * blockDim.x + threadIdx.x;
    if (e >= EE) return;
    float c0 = c2[0], c1 = c2[1];
    int si, di; float g;
    if (e < E) {
        si = src[e]; di = dst[e];
        g = ea[2 * e] * c0 + ea[2 * e + 1] * c1;
    } else {
        si = di = e - E;
        g = ea_mean[2 * si] * c0 + ea_mean[2 * si + 1] * c1;
    }
    float l = s[si] + dsc[di] + g;
    l = (l > 0.f) ? l : 0.2f * l;   // leaky_relu, slope 0.2
    logits[e] = l;
    atomicMax(&mkey[di], enc_f(l));
}

// ---------------- pass 2: p = exp(l - m), segment sum ----------------------
__global__ void edge_expsum_kernel(const int* __restrict__ dst,
                                   const unsigned* __restrict__ mkey,
                                   float* __restrict__ logits,   // in: l, out: p
                                   float* __restrict__ denom,
                                   int E, int EE) {
    int e = blockIdx.x * blockDim.x + threadIdx.x;
    if (e >= EE) return;
    int di = (e < E) ? dst[e] : (e - E);
    float p = __expf(logits[e] - dec_f(mkey[di]));
    logits[e] = p;
    atomicAdd(&denom[di], p);
}

// ---------------- pass 3: agg[dst] += h[src] * alpha -----------------------
__global__ void edge_scatter_kernel(const int* __restrict__ src,
                                    const int* __restrict__ dst,
                                    const float* __restrict__ h,
                                    const float* __restrict__ p,
                                    const float* __restrict__ denom,
                                    float* __restrict__ agg,
                                    int E, int EE) {
    int gid = blockIdx.x * blockDim.x + threadIdx.x;
    int e = gid >> 5;
    int lane = gid & 31;
    if (e >= EE) return;
    int si, di;
    if (e < E) { si = src[e]; di = dst[e]; } else { si = di = e - E; }
    float alpha = p[e] / (denom[di] + 1e-16f);
    const float* hs = h + (size_t)si * 128;
    float* ad = agg + (size_t)di * 128;
#pragma unroll
    for (int j = lane; j < 128; j += 32) atomicAdd(&ad[j], hs[j] * alpha);
}

// ---------------- bias + relu ----------------------------------------------
__global__ void bias_relu_kernel(float* __restrict__ agg,
                                 const float* __restrict__ b, size_t total) {
    size_t i = (size_t)blockIdx.x * blockDim.x + threadIdx.x;
    if (i >= total) return;
    float v = agg[i] + b[i & 127];
    agg[i] = v > 0.f ? v : 0.f;
}

// ---------------- mean pool (partial sums per block) -----------------------
__global__ void pool_kernel(const float* __restrict__ h,
                            float* __restrict__ pooled, int N) {
    int t = threadIdx.x;             // feature column 0..127
    float acc = 0.f;
    for (int r = blockIdx.x; r < N; r += gridDim.x)
        acc += h[(size_t)r * 128 + t];
    atomicAdd(&pooled[t], acc);
}

// ---------------- final head: tanh(pooled/N @ Wfc + bfc) -------------------
__global__ void head_kernel(const float* __restrict__ pooled,
                            const float* __restrict__ Wfc,
                            const float* __restrict__ bfc,
                            float* __restrict__ out, float invN, int A) {
    int a = threadIdx.x >> 5;        // one wave per output
    int lane = threadIdx.x & 31;
    if (a >= A) return;
    float acc = 0.f;
#pragma unroll
    for (int j = lane; j < 128; j += 32)
        acc += pooled[j] * invN * Wfc[(size_t)j * A + a];
#pragma unroll
    for (int o = 16; o > 0; o >>= 1) acc += __shfl_down(acc, o);
    if (lane == 0) out[a] = tanhf(acc + bfc[a]);
}

// ===========================================================================
extern "C" void kernel_launch(void* const* d_in, const int* in_sizes, int n_in,
                              void* d_out, int out_size, void* d_ws, size_t ws_size,
                              hipStream_t stream) {
    const float* x    = (const float*)d_in[0];
    const int*   eidx = (const int*)  d_in[1];
    const float* ea   = (const float*)d_in[2];
    const float* W1   = (const float*)d_in[3];
    const float* We1  = (const float*)d_in[4];
    const float* as1  = (const float*)d_in[5];
    const float* ad1  = (const float*)d_in[6];
    const float* ae1  = (const float*)d_in[7];
    const float* b1   = (const float*)d_in[8];
    const float* W2   = (const float*)d_in[9];
    const float* We2  = (const float*)d_in[10];
    const float* as2  = (const float*)d_in[11];
    const float* ad2  = (const float*)d_in[12];
    const float* ae2  = (const float*)d_in[13];
    const float* b2   = (const float*)d_in[14];
    const float* Wfc  = (const float*)d_in[15];
    const float* bfc  = (const float*)d_in[16];
    float* out = (float*)d_out;

    const int N  = in_sizes[0] / 64;     // 50000
    const int E  = in_sizes[1] / 2;      // 1600000
    const int A  = out_size;             // 16
    const int EE = E + N;                // edges + self loops
    const size_t NH = (size_t)N * 128;

    // ---- workspace carve-out (256B aligned) ----
    char* wsp = (char*)d_ws;
    size_t off = 0;
    auto carve = [&](size_t bytes) -> char* {
        char* p = wsp + off;
        off += (bytes + 255) & ~(size_t)255;
        return p;
    };
    float*    bufA   = (float*)   carve(NH * sizeof(float));         // h (GEMM out)
    float*    bufB   = (float*)   carve(NH * sizeof(float));         // agg / next X
    float*    sbuf   = (float*)   carve((size_t)N * sizeof(float));
    float*    dbuf   = (float*)   carve((size_t)N * sizeof(float));
    unsigned* mkey   = (unsigned*)carve((size_t)N * sizeof(unsigned));
    float*    denom  = (float*)   carve((size_t)N * sizeof(float));
    float*    logits = (float*)   carve((size_t)EE * sizeof(float));
    float*    eamean = (float*)   carve((size_t)N * 2 * sizeof(float));
    float*    cnt    = (float*)   carve((size_t)N * sizeof(float));
    float*    c2buf  = (float*)   carve(64 * sizeof(float));
    float*    pooled = (float*)   carve(128 * sizeof(float));
    (void)ws_size; (void)n_in;

    const int* src = eidx;
    const int* dst = eidx + E;

    const int B256 = 256;
    const int gN    = (N + B256 - 1) / B256;
    const int gE    = (E + B256 - 1) / B256;
    const int gEE   = (EE + B256 - 1) / B256;
    const int gEEw  = (EE + 7) / 8;          // wave-per-edge kernels
    const int gNw   = (N + 7) / 8;           // wave-per-node kernels
    const int gTile = (N + 15) / 16;         // 16-row WMMA tiles
    const int gNH   = (int)((NH + B256 - 1) / B256);

    // ---- self-loop edge_attr mean ----
    hipMemsetAsync(cnt, 0, (size_t)N * sizeof(float), stream);
    hipMemsetAsync(eamean, 0, (size_t)N * 2 * sizeof(float), stream);
    ea_stats_kernel<<<gE, B256, 0, stream>>>(dst, ea, cnt, eamean, E);
    ea_div_kernel<<<gN, B256, 0, stream>>>(eamean, cnt, N);

    // ================= layer 1: F=64 -> H=128 =================
    gemm_wmma_kernel<64><<<gTile, B256, 0, stream>>>(x, W1, bufA, N);
    node_scalars_kernel<<<gNw, B256, 0, stream>>>(bufA, as1, ad1, sbuf, dbuf, N);
    compute_c2_kernel<<<1, 64, 0, stream>>>(We1, ae1, c2buf);
    hipMemsetAsync(mkey, 0, (size_t)N * sizeof(unsigned), stream);
    hipMemsetAsync(denom, 0, (size_t)N * sizeof(float), stream);
    hipMemsetAsync(bufB, 0, NH * sizeof(float), stream);
    edge_logits_max_kernel<<<gEE, B256, 0, stream>>>(src, dst, ea, eamean, c2buf,
                                                     sbuf, dbuf, logits, mkey, E, EE);
    edge_expsum_kernel<<<gEE, B256, 0, stream>>>(dst, mkey, logits, denom, E, EE);
    edge_scatter_kernel<<<gEEw, B256, 0, stream>>>(src, dst, bufA, logits, denom,
                                                   bufB, E, EE);
    bias_relu_kernel<<<gNH, B256, 0, stream>>>(bufB, b1, NH);

    // ================= layer 2: H=128 -> H=128 =================
    gemm_wmma_kernel<128><<<gTile, B256, 0, stream>>>(bufB, W2, bufA, N);
    node_scalars_kernel<<<gNw, B256, 0, stream>>>(bufA, as2, ad2, sbuf, dbuf, N);
    compute_c2_kernel<<<1, 64, 0, stream>>>(We2, ae2, c2buf);
    hipMemsetAsync(mkey, 0, (size_t)N * sizeof(unsigned), stream);
    hipMemsetAsync(denom, 0, (size_t)N * sizeof(float), stream);
    edge_logits_max_kernel<<<gEE, B256, 0, stream>>>(src, dst, ea, eamean, c2buf,
                                                     sbuf, dbuf, logits, mkey, E, EE);
    edge_expsum_kernel<<<gEE, B256, 0, stream>>>(dst, mkey, logits, denom, E, EE);
    hipMemsetAsync(bufB, 0, NH * sizeof(float), stream);   // bufB now free: becomes agg2
    edge_scatter_kernel<<<gEEw, B256, 0, stream>>>(src, dst, bufA, logits, denom,
                                                   bufB, E, EE);
    bias_relu_kernel<<<gNH, B256, 0, stream>>>(bufB, b2, NH);

    // ================= head =================
    hipMemsetAsync(pooled, 0, 128 * sizeof(float), stream);
    pool_kernel<<<256, 128, 0, stream>>>(bufB, pooled, N);
    head_kernel<<<1, 32 * A, 0, stream>>>(pooled, Wfc, bfc, out, 1.0f / (float)N, A);
}